// GlobalGuidedAoERouter_3925600109077
// MI455X (gfx1250) — compile-verified
//
#include <hip/hip_runtime.h>
#include <math.h>

// Problem constants (match reference)
#define BQ 4
#define TQ 256
#define DM 1024
#define NE 16
#define DLQ 64
#define NH 4
#define NTOK (BQ * TQ)   // 1024
#define SEQL (NE + 1)    // 17

typedef __attribute__((ext_vector_type(2))) float v2f;
typedef __attribute__((ext_vector_type(8))) float v8f;

__device__ __forceinline__ float gelu_exact(float x) {
    return 0.5f * x * (1.0f + erff(x * 0.7071067811865475f));
}

// ---------------------------------------------------------------------------
// Kernel 1: per-(b,d) mean of x over T  -> xmean[4][1024]
// ---------------------------------------------------------------------------
__global__ void k_xmean(const float* __restrict__ x, float* __restrict__ xmean) {
    int idx = blockIdx.x * blockDim.x + threadIdx.x;   // 0..4095
    int b = idx >> 10, d = idx & (DM - 1);
    const float* p = x + ((size_t)b * TQ) * DM + d;
    float s = 0.f;
    for (int t = 0; t < TQ; ++t) s += p[(size_t)t * DM];
    xmean[idx] = s * (1.0f / (float)TQ);
}

// ---------------------------------------------------------------------------
// Kernel 2: global path: gctx = xmean@gp_w.T+gp_b ; gb = gelu(gctx@g1.T)@g2.T
// Also zeroes the aux-loss accumulators (runs before k_token every call).
// ---------------------------------------------------------------------------
__global__ void k_global(const float* __restrict__ xmean,
                         const float* __restrict__ gp_w, const float* __restrict__ gp_b,
                         const float* __restrict__ g1_w, const float* __restrict__ g1_b,
                         const float* __restrict__ g2_w, const float* __restrict__ g2_b,
                         float* __restrict__ gctx, float* __restrict__ gb,
                         float* __restrict__ gsum, float* __restrict__ gcnt) {
    __shared__ float s_gctx[BQ][DLQ];
    __shared__ float s_gact[BQ][2 * DLQ];
    int tid = threadIdx.x;

    if (tid < NE) { gsum[tid] = 0.f; gcnt[tid] = 0.f; }

    // phase 1: 4*64 = 256 outputs, one per thread
    {
        int b = tid >> 6, l = tid & 63;
        const float* xm = xmean + (size_t)b * DM;
        const float* w  = gp_w + (size_t)l * DM;
        float acc = gp_b[l];
        for (int d = 0; d < DM; ++d) acc += xm[d] * w[d];
        s_gctx[b][l] = acc;
        gctx[b * DLQ + l] = acc;
    }
    __syncthreads();
    // phase 2: 4*128 = 512 gelu(g1) outputs
    for (int idx = tid; idx < BQ * 2 * DLQ; idx += blockDim.x) {
        int b = idx >> 7, j = idx & 127;
        const float* w = g1_w + (size_t)j * DLQ;
        float acc = g1_b[j];
        for (int l = 0; l < DLQ; ++l) acc += s_gctx[b][l] * w[l];
        s_gact[b][j] = gelu_exact(acc);
    }
    __syncthreads();
    // phase 3: 4*16 = 64 gb outputs
    if (tid < BQ * NE) {
        int b = tid >> 4, e = tid & 15;
        const float* w = g2_w + (size_t)e * (2 * DLQ);
        float acc = g2_b[e];
        for (int j = 0; j < 2 * DLQ; ++j) acc += s_gact[b][j] * w[j];
        gb[b * NE + e] = acc;
    }
}

// ---------------------------------------------------------------------------
// Kernel 3: ef = x(1024x1024) @ w_down.T(1024x1024) via V_WMMA_F32_16X16X4_F32
// 2x2 register-blocked: one wave32 owns a 32x32 output block (4 accumulators).
// Per k-step: 2 A-frag + 2 B-frag loads feed 4 WMMAs -> 8 FLOP/B from L2
// (vs 4 FLOP/B unblocked), halving vmem pressure per WMMA.
// A layout (f32 16x4): lanes 0-15 = M rows with K={0,1}; lanes 16-31 K={2,3}.
// C/D layout: VGPR r, lane L -> M = r + 8*(L/16), N = L%16.
// ---------------------------------------------------------------------------
__global__ void __launch_bounds__(256)
k_down_gemm(const float* __restrict__ xf, const float* __restrict__ wdn,
            float* __restrict__ ef) {
    int wave = threadIdx.x >> 5;
    int lane = threadIdx.x & 31;
    int tile = blockIdx.x * 8 + wave;        // 0..1023  (32x32 blocks of 32x32)
    int row0 = (tile >> 5) << 5;             // token rows   (32-row block)
    int col0 = (tile & 31) << 5;             // ef columns   (32-col block)
    int mr   = lane & 15;
    int koff = (lane >> 4) << 1;             // 0 or 2

    const float* a0p = xf  + (size_t)(row0 + mr)      * DM + koff;
    const float* a1p = xf  + (size_t)(row0 + 16 + mr) * DM + koff;
    const float* b0p = wdn + (size_t)(col0 + mr)      * DM + koff;
    const float* b1p = wdn + (size_t)(col0 + 16 + mr) * DM + koff;

    v8f c00 = {0.f,0.f,0.f,0.f,0.f,0.f,0.f,0.f};
    v8f c01 = c00, c10 = c00, c11 = c00;

#pragma unroll 4
    for (int k = 0; k < DM; k += 4) {
        // pull the next chunk toward the WGP cache (global_prefetch_b8)
        __builtin_prefetch(a0p + k + 64, 0, 1);
        __builtin_prefetch(b0p + k + 64, 0, 1);
        v2f a0 = *(const v2f*)(a0p + k);
        v2f a1 = *(const v2f*)(a1p + k);
        v2f b0 = *(const v2f*)(b0p + k);
        v2f b1 = *(const v2f*)(b1p + k);
        c00 = __builtin_amdgcn_wmma_f32_16x16x4_f32(false, a0, false, b0,
                                                    (short)0, c00, false, false);
        c01 = __builtin_amdgcn_wmma_f32_16x16x4_f32(false, a0, false, b1,
                                                    (short)0, c01, false, false);
        c10 = __builtin_amdgcn_wmma_f32_16x16x4_f32(false, a1, false, b0,
                                                    (short)0, c10, false, false);
        c11 = __builtin_amdgcn_wmma_f32_16x16x4_f32(false, a1, false, b1,
                                                    (short)0, c11, false, false);
    }

    int mhalf = (lane >> 4) << 3;            // 0 or 8
    int nn0 = col0 + (lane & 15);
#pragma unroll
    for (int r = 0; r < 8; ++r) {
        size_t rm0 = (size_t)(row0 + mhalf + r) * (NE * DLQ);
        size_t rm1 = (size_t)(row0 + 16 + mhalf + r) * (NE * DLQ);
        ef[rm0 + nn0]      = c00[r];
        ef[rm0 + nn0 + 16] = c01[r];
        ef[rm1 + nn0]      = c10[r];
        ef[rm1 + nn0 + 16] = c11[r];
    }
}

// ---------------------------------------------------------------------------
// Kernel 4: fully fused per-token pipeline. One block (256 thr) per token.
// ---------------------------------------------------------------------------
__global__ void __launch_bounds__(256)
k_token(const float* __restrict__ ef, const float* __restrict__ pos,
        const float* __restrict__ gctx, const float* __restrict__ gb,
        const float* __restrict__ w_in, const float* __restrict__ b_in,
        const float* __restrict__ w_out, const float* __restrict__ b_out,
        const float* __restrict__ ln_w, const float* __restrict__ ln_b,
        const float* __restrict__ ls_w, const float* __restrict__ ls_b,
        const float* __restrict__ w_up,
        float* __restrict__ out,
        float* __restrict__ gsum, float* __restrict__ gcnt) {
    __shared__ float s_seq[SEQL][DLQ];
    __shared__ float s_q[SEQL][DLQ], s_k[SEQL][DLQ], s_v[SEQL][DLQ];
    __shared__ float s_ctx[SEQL][DLQ];
    __shared__ float s_h[SEQL][DLQ];
    __shared__ float s_mu[SEQL], s_rstd[SEQL];
    __shared__ float s_logits[NE], s_probs[NE];
    __shared__ float s_tw[2];
    __shared__ int   s_ti[2];
    __shared__ float s_act[2][DLQ];

    int n = blockIdx.x;          // token 0..1023
    int b = n >> 8;              // batch (T=256)
    int tid = threadIdx.x;

    // 1) build seq: row 0 = gctx[b]; rows 1..16 = ef[n,e,:] + pos_embed[e,:]
    for (int idx = tid; idx < SEQL * DLQ; idx += blockDim.x) {
        int i = idx / DLQ, l = idx - i * DLQ;
        float v;
        if (i == 0) v = gctx[b * DLQ + l];
        else        v = ef[(size_t)n * (NE * DLQ) + (i - 1) * DLQ + l]
                        + pos[(i - 1) * DLQ + l];
        s_seq[i][l] = v;
    }
    __syncthreads();

    // 2) qkv = seq @ attn_in_w.T + attn_in_b   (17 x 192)
    for (int idx = tid; idx < SEQL * 3 * DLQ; idx += blockDim.x) {
        int i = idx / (3 * DLQ), j = idx - i * (3 * DLQ);
        const float* w = w_in + (size_t)j * DLQ;
        float acc = b_in[j];
        for (int l = 0; l < DLQ; ++l) acc += s_seq[i][l] * w[l];
        if      (j < DLQ)     s_q[i][j] = acc;
        else if (j < 2 * DLQ) s_k[i][j - DLQ] = acc;
        else                  s_v[i][j - 2 * DLQ] = acc;
    }
    __syncthreads();

    // 3) attention: thread (h,i) owns one query row; hd=16, scale=1/sqrt(16)
    if (tid < NH * SEQL) {
        int h = tid / SEQL, i = tid - h * SEQL;
        const float* qi = &s_q[i][h * 16];
        float sc[SEQL];
        float mx = -3.0e38f;
        for (int kk = 0; kk < SEQL; ++kk) {
            const float* kv = &s_k[kk][h * 16];
            float d = 0.f;
            for (int dd = 0; dd < 16; ++dd) d += qi[dd] * kv[dd];
            d *= 0.25f;
            sc[kk] = d;
            mx = fmaxf(mx, d);
        }
        float ssum = 0.f;
        for (int kk = 0; kk < SEQL; ++kk) { sc[kk] = expf(sc[kk] - mx); ssum += sc[kk]; }
        float inv = 1.f / ssum;
        for (int dd = 0; dd < 16; ++dd) {
            float acc = 0.f;
            for (int kk = 0; kk < SEQL; ++kk) acc += sc[kk] * s_v[kk][h * 16 + dd];
            s_ctx[i][h * 16 + dd] = acc * inv;
        }
    }
    __syncthreads();

    // 4) attn_out proj + residual -> s_h
    for (int idx = tid; idx < SEQL * DLQ; idx += blockDim.x) {
        int i = idx / DLQ, j = idx - i * DLQ;
        const float* w = w_out + (size_t)j * DLQ;
        float acc = b_out[j];
        for (int l = 0; l < DLQ; ++l) acc += s_ctx[i][l] * w[l];
        s_h[i][j] = acc + s_seq[i][j];
    }
    __syncthreads();

    // LayerNorm stats per row
    if (tid < SEQL) {
        float mu = 0.f;
        for (int j = 0; j < DLQ; ++j) mu += s_h[tid][j];
        mu *= (1.0f / DLQ);
        float var = 0.f;
        for (int j = 0; j < DLQ; ++j) { float d = s_h[tid][j] - mu; var += d * d; }
        var *= (1.0f / DLQ);
        s_mu[tid] = mu;
        s_rstd[tid] = rsqrtf(var + 1e-5f);
    }
    __syncthreads();
    for (int idx = tid; idx < SEQL * DLQ; idx += blockDim.x) {
        int i = idx / DLQ, j = idx - i * DLQ;
        s_h[i][j] = (s_h[i][j] - s_mu[i]) * s_rstd[i] * ln_w[j] + ln_b[j];
    }
    __syncthreads();

    // 5) router logits over refined rows (1..16)
    if (tid < NE) {
        float lg = ls_b[0];
        for (int l = 0; l < DLQ; ++l) lg += s_h[1 + tid][l] * ls_w[l];
        s_logits[tid] = lg + gb[b * NE + tid];
    }
    __syncthreads();

    // 6) softmax over experts + top-2 (lower index wins ties, like lax.top_k)
    if (tid == 0) {
        float mx = -3.0e38f;
        for (int e = 0; e < NE; ++e) mx = fmaxf(mx, s_logits[e]);
        float sm = 0.f;
        for (int e = 0; e < NE; ++e) { float p = expf(s_logits[e] - mx); s_probs[e] = p; sm += p; }
        float inv = 1.f / sm;
        for (int e = 0; e < NE; ++e) s_probs[e] *= inv;
        int i0 = 0;
        for (int e = 1; e < NE; ++e) if (s_probs[e] > s_probs[i0]) i0 = e;
        int i1 = (i0 == 0) ? 1 : 0;
        for (int e = 0; e < NE; ++e)
            if (e != i0 && s_probs[e] > s_probs[i1]) i1 = e;
        float w0 = s_probs[i0], w1 = s_probs[i1], s = w0 + w1;
        s_ti[0] = i0; s_ti[1] = i1;
        s_tw[0] = w0 / s; s_tw[1] = w1 / s;
    }
    __syncthreads();

    // aux-loss accumulators
    if (tid < NE) atomicAdd(&gsum[tid], s_probs[tid]);
    if (tid < 2)  atomicAdd(&gcnt[s_ti[tid]], 1.0f);

    // 7) act = gelu(ef[n, top_e, :])   (raw ef, no pos_embed — per reference)
    if (tid < 2 * DLQ) {
        int kk = tid >> 6, l = tid & 63;
        float x = ef[(size_t)n * (NE * DLQ) + s_ti[kk] * DLQ + l];
        s_act[kk][l] = gelu_exact(x);
    }
    __syncthreads();

    // 8) out[n, m] = sum_k tw[k] * (act_k @ w_up[e_k])   (m = 0..1023)
    for (int m = tid; m < DM; m += blockDim.x) {
        float acc = 0.f;
#pragma unroll
        for (int kk = 0; kk < 2; ++kk) {
            const float* wu = w_up + ((size_t)s_ti[kk] * DLQ) * DM + m;
            float a = 0.f;
            for (int l = 0; l < DLQ; ++l) a += s_act[kk][l] * wu[(size_t)l * DM];
            acc += s_tw[kk] * a;
        }
        out[(size_t)n * DM + m] = acc;
    }
}

// ---------------------------------------------------------------------------
// Kernel 5: aux_loss = E/N^2 * sum_e sumP[e]*cnt[e]
// ---------------------------------------------------------------------------
__global__ void k_aux(const float* __restrict__ gsum, const float* __restrict__ gcnt,
                      float* __restrict__ aux_out) {
    if (threadIdx.x == 0 && blockIdx.x == 0) {
        float a = 0.f;
        for (int e = 0; e < NE; ++e) a += gsum[e] * gcnt[e];
        aux_out[0] = (float)NE * a / ((float)NTOK * (float)NTOK);
    }
}

// ---------------------------------------------------------------------------
extern "C" void kernel_launch(void* const* d_in, const int* in_sizes, int n_in,
                              void* d_out, int out_size, void* d_ws, size_t ws_size,
                              hipStream_t stream) {
    (void)in_sizes; (void)n_in; (void)out_size; (void)ws_size;
    const float* x          = (const float*)d_in[0];
    const float* w_down     = (const float*)d_in[1];
    const float* pos_embed  = (const float*)d_in[2];
    const float* gp_w       = (const float*)d_in[3];
    const float* gp_b       = (const float*)d_in[4];
    const float* attn_in_w  = (const float*)d_in[5];
    const float* attn_in_b  = (const float*)d_in[6];
    const float* attn_out_w = (const float*)d_in[7];
    const float* attn_out_b = (const float*)d_in[8];
    const float* ln_w       = (const float*)d_in[9];
    const float* ln_b       = (const float*)d_in[10];
    const float* ls_w       = (const float*)d_in[11];
    const float* ls_b       = (const float*)d_in[12];
    const float* g1_w       = (const float*)d_in[13];
    const float* g1_b       = (const float*)d_in[14];
    const float* g2_w       = (const float*)d_in[15];
    const float* g2_b       = (const float*)d_in[16];
    const float* w_up       = (const float*)d_in[17];

    float* out = (float*)d_out;                       // [1024*1024] + aux at end
    float* aux = out + (size_t)NTOK * DM;

    // workspace layout (floats)
    float* ws    = (float*)d_ws;
    float* ef    = ws;                                // 1024*1024
    float* xmean = ef + (size_t)NTOK * DM;            // 4096
    float* gctx  = xmean + BQ * DM;                   // 256
    float* gb    = gctx + BQ * DLQ;                   // 64
    float* gsum  = gb + BQ * NE;                      // 16
    float* gcnt  = gsum + NE;                         // 16

    k_xmean<<<16, 256, 0, stream>>>(x, xmean);
    k_global<<<1, 256, 0, stream>>>(xmean, gp_w, gp_b, g1_w, g1_b, g2_w, g2_b,
                                    gctx, gb, gsum, gcnt);
    k_down_gemm<<<128, 256, 0, stream>>>(x, w_down, ef);
    k_token<<<NTOK, 256, 0, stream>>>(ef, pos_embed, gctx, gb,
                                      attn_in_w, attn_in_b, attn_out_w, attn_out_b,
                                      ln_w, ln_b, ls_w, ls_b, w_up,
                                      out, gsum, gcnt);
    k_aux<<<1, 32, 0, stream>>>(gsum, gcnt, aux);
}